// WaveLayer_20083267076499
// MI455X (gfx1250) — compile-verified
//
#include <hip/hip_runtime.h>

typedef float v2f __attribute__((ext_vector_type(2)));
typedef float v8f __attribute__((ext_vector_type(8)));

#define NXD 128
#define NYD 128
#define TSTEPS 256
#define BATCHD 8
// denom = dt^-2 + 0.5*b/dt = 4.005 ; coef2 = -(1 + dt*b) = -1.0025 ; dt^2 = 0.25

__global__ __launch_bounds__(1024) void wave_sim(const float* __restrict__ x,
                                                 const float* __restrict__ c,
                                                 const long long* __restrict__ probes,
                                                 float* __restrict__ ws_sums)
{
    extern __shared__ float lds[];
    float* buf0 = lds;            // field buffer 0 (16384 floats)
    float* buf1 = lds + NXD*NYD;  // field buffer 1

    const int tid   = threadIdx.x;
    const int lane  = tid & 31;          // wave32
    const int wave  = tid >> 5;          // 32 waves
    const int batch = blockIdx.x;

    // zero both time levels (y0 = 0)
    for (int i = tid; i < 2*NXD*NYD; i += 1024) lds[i] = 0.0f;

    // Tridiagonal T = tridiag(1,-2,1), 16x16, in WMMA f32 chunk layout.
    // Symmetric => same registers serve as A-operand chunks and B-operand chunks.
    // layout[v][lane]: row = lane&15, col = 4*(v>>1) + (v&1) + 2*(lane>>4)
    float T8[8];
    {
        const int r = lane & 15;
        #pragma unroll
        for (int v = 0; v < 8; ++v) {
            const int col = 4*(v>>1) + (v&1) + 2*(lane>>4);
            const int d = r - col;
            T8[v] = (d == 0) ? -2.0f : ((d == 1 || d == -1) ? 1.0f : 0.0f);
        }
    }

    // two 16x16 tiles per wave: tile ids wave and wave+32 ; I = tile/8, J = tile%8
    const int I0 = wave >> 3,        J0 = wave & 7;
    const int I1 = (wave + 32) >> 3, J1 = (wave + 32) & 7;

    // cache dt^2 * c^2 at this lane's C/D-layout positions for both tiles
    float csq0[8], csq1[8];
    {
        const int colL = lane & 15;
        const int rAdd = 8*(lane>>4);
        #pragma unroll
        for (int v = 0; v < 8; ++v) {
            int gr = 16*I0 + v + rAdd, gc = 16*J0 + colL;
            float cv = c[gr*NYD + gc];
            csq0[v] = 0.25f * cv * cv;
            gr = 16*I1 + v + rAdd; gc = 16*J1 + colL;
            cv = c[gr*NYD + gc];
            csq1[v] = 0.25f * cv * cv;
        }
    }

    // probe coordinates: wave 0, lanes 0..3 (probes are int64 pairs)
    int ppx = 0, ppy = 0;
    float pacc = 0.0f;
    if (wave == 0 && lane < 4) {
        ppx = (int)probes[lane*2 + 0];
        ppy = (int)probes[lane*2 + 1];
    }

    __syncthreads();

    float* prev = buf0;   // y1 (t-1)
    float* curr = buf1;   // y2 (t-2), overwritten with y(t)

    const float inv_denom = 1.0f / 4.005f;
    const float coef2     = -1.0025f;

    for (int t = 0; t < TSTEPS; ++t) {
        const float xv = x[t*BATCHD + batch];   // wave-uniform -> s_load

        #pragma unroll
        for (int which = 0; which < 2; ++which) {
            const int I = which ? I1 : I0;
            const int J = which ? J1 : J0;
            const float* csq = which ? csq1 : csq0;
            const int base = (16*I)*NYD + 16*J;

            // Y1 tile: B-operand layout (Yb) and A-operand layout (Ya)
            float Yb[8], Ya[8];
            #pragma unroll
            for (int v = 0; v < 8; ++v) {
                const int kk = 4*(v>>1) + (v&1) + 2*(lane>>4);
                Yb[v] = prev[base + kk*NYD + (lane & 15)];        // row kk, col lane
                Ya[v] = prev[base + (lane & 15)*NYD + kk];        // row lane, col kk
            }

            // lap_tile = T_mid * Y + Y * T_mid  via 8 chained f32 WMMAs (K=16 as 4xK=4)
            v8f lap = {0.f,0.f,0.f,0.f,0.f,0.f,0.f,0.f};
            #pragma unroll
            for (int k = 0; k < 4; ++k) {
                v2f a  = { T8[2*k], T8[2*k+1] };
                v2f bm = { Yb[2*k], Yb[2*k+1] };
                lap = __builtin_amdgcn_wmma_f32_16x16x4_f32(false, a, false, bm,
                                                            (short)0, lap, false, false);
            }
            #pragma unroll
            for (int k = 0; k < 4; ++k) {
                v2f a  = { Ya[2*k], Ya[2*k+1] };
                v2f bm = { T8[2*k], T8[2*k+1] };
                lap = __builtin_amdgcn_wmma_f32_16x16x4_f32(false, a, false, bm,
                                                            (short)0, lap, false, false);
            }

            float lap8[8];
            #pragma unroll
            for (int v = 0; v < 8; ++v) lap8[v] = lap[v];

            // cross-tile boundary strips (zero padding outside the 128x128 field)
            if (lane < 16) {
                if (I > 0) lap8[0] += prev[(16*I - 1)*NYD + 16*J + lane];
            } else {
                if (I < 7) lap8[7] += prev[(16*I + 16)*NYD + 16*J + (lane - 16)];
            }
            if ((lane & 15) == 0 && J > 0) {
                #pragma unroll
                for (int v = 0; v < 8; ++v)
                    lap8[v] += prev[(16*I + v + 8*(lane>>4))*NYD + 16*J - 1];
            }
            if ((lane & 15) == 15 && J < 7) {
                #pragma unroll
                for (int v = 0; v < 8; ++v)
                    lap8[v] += prev[(16*I + v + 8*(lane>>4))*NYD + 16*J + 16];
            }

            // y = (2.0 + dt^2 c^2 lap - (1+dt b) y2) / denom  (+ source), y2 = curr
            #pragma unroll
            for (int v = 0; v < 8; ++v) {
                const int idx = base + (v + 8*(lane>>4))*NYD + (lane & 15);
                const float y2v = curr[idx];
                float yn = (2.0f + csq[v]*lap8[v] + coef2*y2v) * inv_denom;
                if (I == 4 && J == 4 && v == 0 && lane == 0) yn += xv; // (64,64)
                curr[idx] = yn;
            }
        }

        __syncthreads();   // publish y(t) before anyone reads it as prev / probes

        // probe accumulation on the just-written field
        if (wave == 0 && lane < 4) {
            const float pv = curr[ppx*NYD + ppy];
            pacc += pv * pv;
        }

        // rotate: (y1, y2) <- (y, y1)
        float* tmp = prev; prev = curr; curr = tmp;
    }

    if (wave == 0 && lane < 4) atomicAdd(&ws_sums[lane], pacc);
}

__global__ void zero4(float* p) {
    if (threadIdx.x < 4) p[threadIdx.x] = 0.0f;
}

__global__ void finalize4(const float* __restrict__ s, float* __restrict__ out) {
    if (threadIdx.x < 4) {
        const float tot = s[0] + s[1] + s[2] + s[3];
        out[threadIdx.x] = s[threadIdx.x] / tot;
    }
}

extern "C" void kernel_launch(void* const* d_in, const int* in_sizes, int n_in,
                              void* d_out, int out_size, void* d_ws, size_t ws_size,
                              hipStream_t stream) {
    (void)in_sizes; (void)n_in; (void)out_size; (void)ws_size;
    const float*     x      = (const float*)d_in[0];       // (256, 8) f32
    const float*     c      = (const float*)d_in[1];       // (128, 128) f32
    const long long* probes = (const long long*)d_in[2];   // (4, 2) int64
    float* out  = (float*)d_out;                           // 4 f32
    float* sums = (float*)d_ws;                            // 4-float accumulator

    zero4<<<1, 32, 0, stream>>>(sums);
    const size_t shmem = (size_t)(2 * NXD * NYD) * sizeof(float);  // 128 KB < 320 KB/WGP
    wave_sim<<<BATCHD, 1024, shmem, stream>>>(x, c, probes, sums);
    finalize4<<<1, 32, 0, stream>>>(sums, out);
}